// GAT_39256001085807
// MI455X (gfx1250) — compile-verified
//
#include <hip/hip_runtime.h>
#include <math.h>
#include <stdint.h>

#define N_NODES 50000

typedef __attribute__((ext_vector_type(2))) float v2f;
typedef __attribute__((ext_vector_type(8))) float v8f;

// explicit global-address-space pointer types (avoid flat_load codegen).
// NOTE: use ext_vector (trivial) types, not HIP_vector_type classes, so
// loads through addrspace(1) lvalues are plain copies.
typedef const __attribute__((address_space(1))) float gcf;
typedef const __attribute__((address_space(1))) v2f   gcv2;
typedef __attribute__((address_space(1))) float       gf;

__device__ __forceinline__ const gcf* as_global(const float* p) {
    return (const gcf*)(uintptr_t)p;
}
__device__ __forceinline__ gf* as_global_mut(float* p) {
    return (gf*)(uintptr_t)p;
}

// ---- ordered-int encoding for float atomicMax -------------------------------
__device__ __forceinline__ unsigned f2ord(float f) {
    unsigned u = __float_as_uint(f);
    return (u & 0x80000000u) ? ~u : (u | 0x80000000u);
}
__device__ __forceinline__ float ord2f(unsigned u) {
    return (u & 0x80000000u) ? __uint_as_float(u & 0x7fffffffu)
                             : __uint_as_float(~u);
}
#define ORD_NEG_INF 0x007FFFFFu   // f2ord(-inf)

// ---- WMMA f32 GEMM: C[M,N] = A[M,K] @ B[K,N], row-major ---------------------
// One wave computes an (MT*16) x (NT*16) strip via V_WMMA_F32_16X16X4_F32.
// Per K-step: MT A-frags + NT B-frags feed MT*NT WMMAs.  Next K-step's frags
// are prefetched into registers before the current WMMAs (software pipeline).
// All memory accesses via explicit addrspace(1) pointers -> global_load only.
template <int MT, int NT, bool GUARD>
__global__ void wmma_gemm_f32(const float* __restrict__ A,
                              const float* __restrict__ B,
                              float* __restrict__ C,
                              int M, int K, int N, int mgroups, int ngroups) {
    const int wid  = (blockIdx.x * blockDim.x + threadIdx.x) >> 5;
    const int lane = threadIdx.x & 31;
    if (wid >= mgroups * ngroups) return;         // wave-uniform exit
    const int mg   = wid / ngroups;
    const int ng   = wid % ngroups;
    const int half = lane >> 4;                   // 0: lanes 0-15, 1: lanes 16-31
    const int ln   = lane & 15;

    // A frags: M = lane index; lanes 0-15 hold K={k,k+1}, 16-31 {k+2,k+3}
    const gcf* Ap[MT];
    int rowm[MT];
    #pragma unroll
    for (int m = 0; m < MT; ++m) {
        const int r  = mg * MT * 16 + m * 16 + ln;
        rowm[m] = mg * MT * 16 + m * 16;          // store base (lane-invariant part)
        const int rc = (r < M) ? r : (M - 1);     // clamp tail rows (load-safe)
        Ap[m] = as_global(A) + (size_t)rc * K + 2 * half;
    }

    int        coln[NT];
    bool       inn[NT];
    const gcf* Bp[NT];
    #pragma unroll
    for (int t = 0; t < NT; ++t) {
        const int cb = (ng * NT + t) * 16 + ln;
        coln[t] = cb;
        inn[t]  = !GUARD || (cb < N);
        const int cc = GUARD ? (cb < N ? cb : N - 1) : cb;
        Bp[t] = as_global(B) + (size_t)(2 * half) * N + cc;  // row k0+2*half, col cc
    }

    v8f acc[MT][NT] = {};
    const size_t bstep = (size_t)4 * N;

    v2f aR[MT], bR[NT], aN[MT], bN[NT];
    // prologue: load K-step 0
    #pragma unroll
    for (int m = 0; m < MT; ++m) aR[m] = *(const gcv2*)(Ap[m]);
    #pragma unroll
    for (int t = 0; t < NT; ++t) {
        const float b0 = Bp[t][0], b1 = Bp[t][N];
        bR[t][0] = (!GUARD || inn[t]) ? b0 : 0.0f;
        bR[t][1] = (!GUARD || inn[t]) ? b1 : 0.0f;
        Bp[t] += bstep;
    }

    for (int k0 = 0; k0 < K - 4; k0 += 4) {
        // prefetch next K-step
        #pragma unroll
        for (int m = 0; m < MT; ++m) aN[m] = *(const gcv2*)(Ap[m] + k0 + 4);
        #pragma unroll
        for (int t = 0; t < NT; ++t) {
            const float b0 = Bp[t][0], b1 = Bp[t][N];
            bN[t][0] = (!GUARD || inn[t]) ? b0 : 0.0f;
            bN[t][1] = (!GUARD || inn[t]) ? b1 : 0.0f;
            Bp[t] += bstep;
        }
        // compute current K-step
        #pragma unroll
        for (int m = 0; m < MT; ++m) {
            #pragma unroll
            for (int t = 0; t < NT; ++t) {
                acc[m][t] = __builtin_amdgcn_wmma_f32_16x16x4_f32(
                                false, aR[m], false, bR[t], (short)0, acc[m][t],
                                false, false);
            }
        }
        #pragma unroll
        for (int m = 0; m < MT; ++m) aR[m] = aN[m];
        #pragma unroll
        for (int t = 0; t < NT; ++t) bR[t] = bN[t];
    }
    // epilogue: last K-step
    #pragma unroll
    for (int m = 0; m < MT; ++m) {
        #pragma unroll
        for (int t = 0; t < NT; ++t) {
            acc[m][t] = __builtin_amdgcn_wmma_f32_16x16x4_f32(
                            false, aR[m], false, bR[t], (short)0, acc[m][t],
                            false, false);
        }
    }

    gf* Cg = as_global_mut(C);
    #pragma unroll
    for (int m = 0; m < MT; ++m) {
        #pragma unroll
        for (int t = 0; t < NT; ++t) {
            if (inn[t]) {
                #pragma unroll
                for (int r = 0; r < 8; ++r) {
                    const int row = rowm[m] + r + 8 * half;   // C: VGPR r -> M=r (+8 hi)
                    if (MT == 1 || row < M)
                        Cg[(size_t)row * N + coln[t]] = acc[m][t][r];
                }
            }
        }
    }
}

// ---- per-(node,head) attention coefficients a_s, a_d ------------------------
__global__ void attn_coef(const float* __restrict__ h,
                          const float* __restrict__ att_s,
                          const float* __restrict__ att_d,
                          float* __restrict__ a_s, float* __restrict__ a_d,
                          int H, int C) {
    const int tid = blockIdx.x * blockDim.x + threadIdx.x;
    if (tid >= N_NODES * H) return;
    const int n = tid / H, hh = tid % H;
    const float4* hp = (const float4*)(h + (size_t)n * H * C + hh * C);
    const float4* sp = (const float4*)(att_s + hh * C);
    const float4* dp = (const float4*)(att_d + hh * C);
    float ss = 0.0f, dd = 0.0f;
    for (int i = 0; i < (C >> 2); ++i) {
        const float4 hv = hp[i], sv = sp[i], dv = dp[i];
        ss += hv.x * sv.x + hv.y * sv.y + hv.z * sv.z + hv.w * sv.w;
        dd += hv.x * dv.x + hv.y * dv.y + hv.z * dv.z + hv.w * dv.w;
    }
    a_s[tid] = ss;
    a_d[tid] = dd;
}

// ---- init accumulators ------------------------------------------------------
__global__ void init_attn(float* __restrict__ acc, float* __restrict__ denom,
                          unsigned* __restrict__ mbuf, int n_acc, int n_head) {
    const int tid = blockIdx.x * blockDim.x + threadIdx.x;
    if (tid < n_acc) acc[tid] = 0.0f;
    if (tid < n_head) { denom[tid] = 0.0f; mbuf[tid] = ORD_NEG_INF; }
}

// ---- edge pass 1: scatter-max of leaky-relu logits --------------------------
__global__ void edge_max(const int* __restrict__ ei, int E, int Etot,
                         const float* __restrict__ a_s,
                         const float* __restrict__ a_d,
                         unsigned* __restrict__ mbuf, int H) {
    const int tid = blockIdx.x * blockDim.x + threadIdx.x;
    if (tid >= Etot * H) return;
    const int e = tid / H, hh = tid % H;
    const int s = (e < E) ? ei[e]     : (e - E);   // self-loops appended
    const int d = (e < E) ? ei[E + e] : (e - E);
    float x = a_s[s * H + hh] + a_d[d * H + hh];
    x = (x > 0.0f) ? x : 0.2f * x;                 // leaky_relu(0.2)
    atomicMax(&mbuf[d * H + hh], f2ord(x));
}

// ---- edge pass 2: unnormalized softmax-weighted scatter-add -----------------
// One wave per edge; each lane owns 4 contiguous channels (float4 gather,
// C % 4 == 0 so a 4-channel group never straddles heads).
__global__ void edge_accum(const int* __restrict__ ei, int E, int Etot,
                           const float* __restrict__ a_s,
                           const float* __restrict__ a_d,
                           const unsigned* __restrict__ mbuf,
                           const float* __restrict__ hfeat,
                           float* __restrict__ acc, float* __restrict__ denom,
                           int H, int C) {
    const int wid  = (blockIdx.x * blockDim.x + threadIdx.x) >> 5;
    const int lane = threadIdx.x & 31;
    if (wid >= Etot) return;
    const int s = (wid < E) ? ei[wid]     : (wid - E);
    const int d = (wid < E) ? ei[E + wid] : (wid - E);
    const int HC = H * C;
    const float* hs = hfeat + (size_t)s * HC;
    float*       ad = acc   + (size_t)d * HC;
    for (int base = lane * 4; base < HC; base += 128) {
        const int hh = base / C;
        float x = a_s[s * H + hh] + a_d[d * H + hh];
        x = (x > 0.0f) ? x : 0.2f * x;
        const float w = expf(x - ord2f(mbuf[d * H + hh]));
        const float4 hv = *(const float4*)(hs + base);
        atomicAdd(&ad[base + 0], w * hv.x);
        atomicAdd(&ad[base + 1], w * hv.y);
        atomicAdd(&ad[base + 2], w * hv.z);
        atomicAdd(&ad[base + 3], w * hv.w);
        if ((base % C) == 0) atomicAdd(&denom[d * H + hh], w);
    }
}

// ---- finalize: normalize, bias, optional ELU --------------------------------
__global__ void finalize(const float* __restrict__ acc,
                         const float* __restrict__ denom,
                         const float* __restrict__ bias,
                         float* __restrict__ out, int H, int C, int do_elu) {
    const int tid = blockIdx.x * blockDim.x + threadIdx.x;
    const int HC = H * C;
    if (tid >= N_NODES * HC) return;
    const int n = tid / HC, ch = tid % HC;
    float v = acc[tid] / denom[n * H + ch / C] + bias[ch];
    if (do_elu) v = (v > 0.0f) ? v : (expf(v) - 1.0f);
    out[tid] = v;
}

// -----------------------------------------------------------------------------
extern "C" void kernel_launch(void* const* d_in, const int* in_sizes, int n_in,
                              void* d_out, int out_size, void* d_ws, size_t ws_size,
                              hipStream_t stream) {
    const float* x   = (const float*)d_in[0];
    const int*   ei  = (const int*)  d_in[1];
    const float* W1  = (const float*)d_in[2];
    const float* as1 = (const float*)d_in[3];
    const float* ad1 = (const float*)d_in[4];
    const float* b1  = (const float*)d_in[5];
    const float* W2  = (const float*)d_in[6];
    const float* as2 = (const float*)d_in[7];
    const float* ad2 = (const float*)d_in[8];
    const float* b2  = (const float*)d_in[9];
    const float* W3  = (const float*)d_in[10];
    const float* as3 = (const float*)d_in[11];
    const float* ad3 = (const float*)d_in[12];
    const float* b3  = (const float*)d_in[13];

    const int E    = in_sizes[1] / 2;          // 800000
    const int Etot = E + N_NODES;              // + self loops
    const int F_IN = in_sizes[0] / N_NODES;    // 512
    const int H    = 4, C = 64, D1 = H * C;    // 256
    const int NCLS = in_sizes[13];             // 40

    // workspace layout (floats)
    float* ws  = (float*)d_ws;
    const size_t NHC = (size_t)N_NODES * D1;   // 12.8M
    float*    hA  = ws;
    float*    hB  = ws + NHC;
    float*    acc = ws + 2 * NHC;
    float*    aS  = ws + 3 * NHC;
    float*    aD  = aS + (size_t)N_NODES * H;
    float*    den = aD + (size_t)N_NODES * H;
    unsigned* mb  = (unsigned*)(den + (size_t)N_NODES * H);

    const int BLK = 256;
    auto cdiv = [](long long a, long long b) { return (int)((a + b - 1) / b); };

    // per-layer launcher for the edge phase
    auto run_attn = [&](const float* hfeat, const float* atts, const float* attd,
                        const float* bias, float* out_ptr, int Hh, int Cc, int elu) {
        const int HC = Hh * Cc;
        attn_coef<<<cdiv((long long)N_NODES * Hh, BLK), BLK, 0, stream>>>(
            hfeat, atts, attd, aS, aD, Hh, Cc);
        init_attn<<<cdiv((long long)N_NODES * HC, BLK), BLK, 0, stream>>>(
            acc, den, mb, N_NODES * HC, N_NODES * Hh);
        edge_max<<<cdiv((long long)Etot * Hh, BLK), BLK, 0, stream>>>(
            ei, E, Etot, aS, aD, mb, Hh);
        edge_accum<<<cdiv((long long)Etot * 32, BLK), BLK, 0, stream>>>(
            ei, E, Etot, aS, aD, mb, hfeat, acc, den, Hh, Cc);
        finalize<<<cdiv((long long)N_NODES * HC, BLK), BLK, 0, stream>>>(
            acc, den, bias, out_ptr, Hh, Cc, elu);
    };

    auto run_gemm = [&](const float* A, const float* B, float* Cm, int K, int N) {
        const int mtiles = N_NODES / 16;       // 3125
        if ((N & 63) == 0) {
            // 32x64 strip per wave: MT=2, NT=4 (tail M-tile clamped)
            const int mgroups = (mtiles + 1) / 2;
            const int ngroups = N >> 6;
            const long long waves = (long long)mgroups * ngroups;
            wmma_gemm_f32<2, 4, false><<<cdiv(waves * 32, BLK), BLK, 0, stream>>>(
                A, B, Cm, N_NODES, K, N, mgroups, ngroups);
        } else {
            // guarded single-tile path (N = 40 tail)
            const int ngroups = (N + 15) / 16;
            const long long waves = (long long)mtiles * ngroups;
            wmma_gemm_f32<1, 1, true><<<cdiv(waves * 32, BLK), BLK, 0, stream>>>(
                A, B, Cm, N_NODES, K, N, mtiles, ngroups);
        }
    };

    // Layer 1: x[N,512] @ W1 -> hA[N,256]; attn -> hB (ELU)
    run_gemm(x, W1, hA, F_IN, D1);
    run_attn(hA, as1, ad1, b1, hB, H, C, 1);

    // Layer 2: hB @ W2 -> hA; attn -> hB (ELU)
    run_gemm(hB, W2, hA, D1, D1);
    run_attn(hA, as2, ad2, b2, hB, H, C, 1);

    // Layer 3: hB @ W3 -> hA[N,40]; attn (H=1,C=40) -> d_out (no ELU)
    run_gemm(hB, W3, hA, D1, NCLS);
    run_attn(hA, as3, ad3, b3, (float*)d_out, 1, NCLS, 0);
}